// ConvEnhanced_65481071410106
// MI455X (gfx1250) — compile-verified
//
#include <hip/hip_runtime.h>
#include <hip/hip_bf16.h>
#include <math.h>

typedef __attribute__((ext_vector_type(2))) float v2f;
typedef __attribute__((ext_vector_type(8))) float v8f;

// ---------------------------------------------------------------------------
// Kernel 1: simulate the 4-qubit circuit, write scalar to ws[0].
// Lane-0-only with uniform inputs -> compiler scalarizes onto the SALU.
// State layout matches jnp (2,2,2,2) row-major: wire w <-> bit (3-w).
// RX(t)=[[c,-is],[-is,c]], RY(t)=[[c,-s],[s,c]], c=cos(t/2), s=sin(t/2).
// CNOT(c,t): swap amplitudes with ctrl bit = 1 across the target bit.
// Answer = sum_idx |amp|^2 * (4 - popcount(idx)) / 4.
// ---------------------------------------------------------------------------
__global__ void qsim4_scalar_kernel(const float* __restrict__ params,
                                    float* __restrict__ ws) {
  if (threadIdx.x != 0 || blockIdx.x != 0) return;

  float sr[16], si[16];
#pragma unroll
  for (int k = 0; k < 16; ++k) { sr[k] = 0.0f; si[k] = 0.0f; }
  sr[0] = 1.0f;

#pragma unroll
  for (int w = 0; w < 4; ++w) {
    const int str = 1 << (3 - w);

    // RX(params[w])
    float c = cosf(params[w] * 0.5f);
    float s = sinf(params[w] * 0.5f);
#pragma unroll
    for (int idx = 0; idx < 16; ++idx) {
      if (idx & str) continue;
      const int lo = idx, hi = idx | str;
      const float lr = sr[lo], li = si[lo];
      const float hr = sr[hi], hm = si[hi];
      // new_lo = c*lo + (-i s)*hi ; new_hi = (-i s)*lo + c*hi
      sr[lo] = c * lr + s * hm;  si[lo] = c * li - s * hr;
      sr[hi] = c * hr + s * li;  si[hi] = c * hm - s * lr;
    }

    // RY(params[w+4])
    c = cosf(params[w + 4] * 0.5f);
    s = sinf(params[w + 4] * 0.5f);
#pragma unroll
    for (int idx = 0; idx < 16; ++idx) {
      if (idx & str) continue;
      const int lo = idx, hi = idx | str;
      const float lr = sr[lo], li = si[lo];
      const float hr = sr[hi], hm = si[hi];
      sr[lo] = c * lr - s * hr;  si[lo] = c * li - s * hm;
      sr[hi] = s * lr + c * hr;  si[hi] = s * li + c * hm;
    }
  }

  // CNOT chain: ctrl = wire i (bit 3-i), tgt = wire i+1 (bit 2-i)
#pragma unroll
  for (int i = 0; i < 3; ++i) {
    const int bc = 1 << (3 - i);
    const int bt = 1 << (2 - i);
#pragma unroll
    for (int idx = 0; idx < 16; ++idx) {
      if (!(idx & bc)) continue;
      if (idx & bt) continue;
      const int j = idx | bt;
      const float tr = sr[idx], ti = si[idx];
      sr[idx] = sr[j]; si[idx] = si[j];
      sr[j] = tr;      si[j] = ti;
    }
  }

  float acc = 0.0f;
#pragma unroll
  for (int idx = 0; idx < 16; ++idx) {
    const float p = sr[idx] * sr[idx] + si[idx] * si[idx];
    acc += p * (float)(4 - __popc(idx));
  }
  ws[0] = acc * 0.25f;
}

// ---------------------------------------------------------------------------
// Kernel 2: branch-free broadcast of ws[0] through the matrix pipe.
// Per wave: D(16x16) = A(16x4) x B(4x16), A = all val, B = all 0.25
//   => every D entry = 4 * (val * 0.25) = val, bit-exact (pow2 scaling),
// independent of WMMA operand lane layout (all registers uniform).
// Each lane stores its 8 f32 accumulators as two b128 stores -> 256
// outputs per wave, 2048 per 256-thread block. The grid covers only full
// waves, so EXEC is all-ones for the entire kernel (WMMA requirement).
// ---------------------------------------------------------------------------
__global__ void broadcast_wmma_kernel(const float* __restrict__ ws,
                                      float* __restrict__ out) {
  const int lane = threadIdx.x & 31;
  const int wave = threadIdx.x >> 5;
  const long long base =
      (((long long)blockIdx.x * (blockDim.x >> 5) + wave) << 8) |
      ((long long)lane << 3);

  const float val = ws[0];

  v2f a; a[0] = val;   a[1] = val;
  v2f b; b[0] = 0.25f; b[1] = 0.25f;
  v8f c = {};

  // 8 args: (neg_a, A, neg_b, B, c_mod, C, reuse_a, reuse_b)
  v8f d = __builtin_amdgcn_wmma_f32_16x16x4_f32(
      false, a, false, b, (short)0, c, false, false);

  *(v8f*)(out + base) = d;  // 32B-aligned -> clause of 2x global_store_b128
}

// ---------------------------------------------------------------------------
// Kernel 3: scalar fill for any remainder not covered by full blocks.
// Never launched for out_size = 65536 (2048 | 65536); exists for safety.
// ---------------------------------------------------------------------------
__global__ void tail_fill_kernel(const float* __restrict__ ws,
                                 float* __restrict__ out,
                                 int start, int n) {
  const int i = start + blockIdx.x * blockDim.x + threadIdx.x;
  if (i < n) out[i] = ws[0];
}

extern "C" void kernel_launch(void* const* d_in, const int* in_sizes, int n_in,
                              void* d_out, int out_size, void* d_ws, size_t ws_size,
                              hipStream_t stream) {
  (void)in_sizes; (void)n_in; (void)ws_size;
  const float* params = (const float*)d_in[1];  // d_in[0] = x (unused by circuit)
  float* out = (float*)d_out;
  float* ws  = (float*)d_ws;

  // Stage 1: compute the circuit scalar into ws[0] (one wave, SALU-heavy).
  qsim4_scalar_kernel<<<1, 32, 0, stream>>>(params, ws);

  // Stage 2: WMMA broadcast over all full 2048-element blocks.
  const int fullBlocks = out_size / 2048;          // 32 for out_size = 65536
  if (fullBlocks > 0)
    broadcast_wmma_kernel<<<fullBlocks, 256, 0, stream>>>(ws, out);

  // Stage 3: remainder (none for 65536; deterministic since out_size is fixed).
  const int done = fullBlocks * 2048;
  const int rem  = out_size - done;
  if (rem > 0)
    tail_fill_kernel<<<(rem + 255) / 256, 256, 0, stream>>>(ws, out, done, out_size);
}